// MovingAverage_32650341384222
// MI455X (gfx1250) — compile-verified
//
#include <hip/hip_runtime.h>
#include <stdint.h>

// ---------------------------------------------------------------------------
// MovingAverage (B=16,F=32,T=2048,ORDER=8) for MI455X / gfx1250.
//
//   x_Sigma[b,:,:,i] = sum_k F_k * S_{i+k} * F_k^T      (F_8 == Identity)
//   x_mu  [b,c,i]    = sum_{C,k} a_mu[b,C,i+k] * filtr[c,C,k] + bias[c]
//
// ~38.5 GFLOP f32 vs ~260MB traffic -> compute bound in f32 -> dense
// V_WMMA_F32_16X16X4_F32 sandwich GEMMs. All bulk data movement goes
// through the Tensor Data Mover:
//   - a_Sigma slab  : iterated TDM load (Sp<->C transpose-gather + pad
//                     engine producing the conflict-free 36-DWORD rows)
//   - x_Sigma tile  : TDM store from packed LDS staging (async writeback)
//   - a_mu tile/taps: TDM loads (pad engine -> 129-float rows)
// synchronized with s_wait_tensorcnt + workgroup barriers.
// ---------------------------------------------------------------------------

typedef __attribute__((ext_vector_type(2))) float    v2f;
typedef __attribute__((ext_vector_type(8))) float    v8f;
typedef __attribute__((ext_vector_type(4))) unsigned u32x4;
typedef __attribute__((ext_vector_type(4))) int      i32x4;
typedef __attribute__((ext_vector_type(8))) int      i32x8;

constexpr int BATCH = 16;
constexpr int NF    = 32;            // feature dim
constexpr int TIN   = 2048;
constexpr int ORDER = 8;
constexpr int KTAP  = 9;
constexpr int TOUT  = TIN - ORDER;   // 2040

// ---- sigma kernel tiling ----
constexpr int TT    = 24;            // output t per workgroup (2040 = 85*24)
constexpr int TL    = TT + ORDER;    // 32 t loaded (incl. halo)
constexpr int NW    = 8;             // waves per workgroup (wave32)
constexpr int IPW   = TT / NW;       // 3 outputs per wave
constexpr int SPAD  = 36;            // slab row stride (32 data + 4 pad DWORDs, TDM pad engine)
constexpr int FPAD  = 34;            // filter row stride: 8B-aligned pairs, conflict-free
constexpr int VPAD2 = 34;            // V bounce stride [s][C]: contiguous b64 B-operand pairs
constexpr int OPAD  = TT;            // 24: output staging packed for TENSOR_STORE_FROM_LDS

// LDS layout (low -> high): vstg (hot, fits DS 16-bit offset window), filt, slab
constexpr int VSTG_FLOATS   = NW * NF * VPAD2;      // 8704   (34KB)
constexpr int FILT_FLOATS   = KTAP * NF * FPAD;     // 9792   (38.25KB)
constexpr int S_SLAB_FLOATS = NF * NF * SPAD;       // 36864  (144KB)
constexpr int LDS_FLOATS    = VSTG_FLOATS + FILT_FLOATS + S_SLAB_FLOATS;
constexpr size_t LDS_BYTES  = (size_t)LDS_FLOATS * sizeof(float);
constexpr unsigned SLAB_BYTE_OFF = (unsigned)(VSTG_FLOATS + FILT_FLOATS) * 4u;

static_assert(TOUT % TT == 0, "t tiling must be exact");
static_assert(LDS_BYTES <= 320u * 1024u, "fits in WGP LDS");
static_assert(NF * NF * OPAD <= S_SLAB_FLOATS, "output staging reuses slab");

#define WMMA_F32X4(A, Bm, Cm) \
  __builtin_amdgcn_wmma_f32_16x16x4_f32(false, (A), false, (Bm), (short)0, (Cm), false, false)

// ---------------------------------------------------------------------------
// TDM descriptor builder + issue. All parameters are compile-time constants
// at each call site so the descriptor folds to SGPR immediates.
// ---------------------------------------------------------------------------
template <bool STORE>
__device__ inline void tdm_issue(uint64_t ga, unsigned lds_byte,
                                 unsigned tdim0, unsigned tdim1, uint64_t str0,
                                 unsigned tile0, unsigned tile1,
                                 bool pad_en, unsigned pad_ivl, unsigned pad_amt,
                                 bool it_en, unsigned lds_inc, unsigned g_inc,
                                 unsigned it_cnt) {
  u32x4 g0;
  g0[0] = 0x1u;                                   // count=1, user descriptor
  g0[1] = lds_byte;                               // lds_addr (bytes)
  g0[2] = (unsigned)(ga & 0xffffffffu);           // global_addr[31:0]
  g0[3] = (unsigned)((ga >> 32) & 0x01ffffffu)    // global_addr[56:32]
        | (2u << 30);                             // type=2 ("image")

  unsigned flags = (2u << 16);                    // data_size = 4B
  if (it_en)  flags |= (1u << 19);                // iterate_enable
  if (pad_en) flags |= (1u << 20) | (pad_ivl << 22) | (pad_amt << 25);

  i32x8 g1;
  g1[0] = (int)flags;
  g1[1] = (int)((tdim0 & 0xffffu) << 16);         // abar=0 | tensor_dim0 lo16
  g1[2] = (int)(((tdim0 >> 16) & 0xffffu) | ((tdim1 & 0xffffu) << 16));
  g1[3] = (int)(((tdim1 >> 16) & 0xffffu) | (tile0 << 16));
  g1[4] = (int)tile1;                             // tile_dim1 | tile_dim2=0
  g1[5] = (int)(unsigned)(str0 & 0xffffffffu);    // tensor_dim0_stride lo32
  g1[6] = (int)(unsigned)((str0 >> 32) & 0xffffu);
  g1[7] = 0;

  i32x4 g2;
  g2[0] = 1;                                      // tensor_dim2 (unused)
  g2[1] = it_en ? (int)lds_inc : 1;               // lds_addr_increment (elems)
  g2[2] = it_en ? (int)g_inc : 0;                 // global_addr_increment (elems)
  g2[3] = it_en ? (int)(it_cnt << 16) : 0;        // iterate_count

  i32x4 g3 = {0, 0, 0, 0};

#if defined(__clang_major__) && __clang_major__ >= 23
  i32x8 gx = {0, 0, 0, 0, 0, 0, 0, 0};
  if (STORE) __builtin_amdgcn_tensor_store_from_lds(g0, g1, g2, g3, gx, 0);
  else       __builtin_amdgcn_tensor_load_to_lds(g0, g1, g2, g3, gx, 0);
#else
  if (STORE) __builtin_amdgcn_tensor_store_from_lds(g0, g1, g2, g3, 0);
  else       __builtin_amdgcn_tensor_load_to_lds(g0, g1, g2, g3, 0);
#endif
}

// ---------------------------------------------------------------------------
// x_Sigma kernel: one workgroup = (b, 24 t's). 8 waves, 3 t's per wave.
// ---------------------------------------------------------------------------
__global__ __launch_bounds__(NW * 32)
void sigma_wmma_kernel(const float* __restrict__ aS,
                       const float* __restrict__ ma,
                       float* __restrict__ xSig) {
  extern __shared__ float lds[];
  float* vstg_all = lds;                           // per-wave [s*VPAD2 + C], immediate-offset window
  float* filt     = lds + VSTG_FLOATS;             // [k][row*FPAD + col] = filtr[row,col,k]
  float* s_slab   = lds + VSTG_FLOATS + FILT_FLOATS; // [(Sp*32+C)*SPAD + tt]

  const int tid  = threadIdx.x;
  const int lane = tid & 31;
  const int w    = tid >> 5;
  const int m    = lane & 15;    // row/col within 16-tile
  const int hhi  = lane >> 4;    // half-wave select
  const int t0   = blockIdx.x * TT;
  const int b    = blockIdx.y;
  const unsigned lds0 = __builtin_amdgcn_groupstaticsize();

  // ---- TDM load: whole (1024 x 32t) a_Sigma slab, Sp<->C transpose-gather.
  //   tile x:32 t, y:32 rows advancing by C (stride 32*2048); iterate 32x
  //   over Sp (global += 2048, lds += 32*SPAD); pad +4 DW per 32 DW row.
  if (w == 0) {
    const float* gbase = aS + (size_t)b * NF * NF * TIN + t0;
    tdm_issue<false>((uint64_t)(uintptr_t)gbase, lds0 + SLAB_BYTE_OFF,
                     /*tdim0*/ TIN, /*tdim1*/ NF * NF, /*str0*/ (uint64_t)NF * TIN,
                     /*tile0*/ TL, /*tile1*/ NF,
                     /*pad*/ true, /*ivl: 2^(4+1)=32 DW*/ 4u, /*amt: 4 DW*/ 3u,
                     /*iterate*/ true, /*lds_inc*/ NF * SPAD, /*g_inc*/ TIN,
                     /*count*/ NF - 1u);
  }

  // ---- filter taps -> LDS (identity tap materialized as k=8) ----
  for (int idx = tid; idx < NF * NF * ORDER; idx += NW * 32) {
    int r  = idx >> 8;
    int c2 = (idx >> 3) & 31;
    int k  = idx & 7;                              // coalesced global read of ma
    filt[k * NF * FPAD + r * FPAD + c2] = ma[idx];
  }
  for (int idx = tid; idx < NF * NF; idx += NW * 32) {
    int r = idx >> 5, c2 = idx & 31;
    filt[ORDER * NF * FPAD + r * FPAD + c2] = (r == c2) ? 1.0f : 0.0f;
  }

  if (w == 0) __builtin_amdgcn_s_wait_tensorcnt(0);
  __syncthreads();

  float* vstg = vstg_all + w * NF * VPAD2;

  v8f acc[IPW][4];
#pragma unroll
  for (int il = 0; il < IPW; ++il)
#pragma unroll
    for (int q = 0; q < 4; ++q)
      acc[il][q] = (v8f){0.f, 0.f, 0.f, 0.f, 0.f, 0.f, 0.f, 0.f};

  for (int k = 0; k < ORDER; ++k) {                // 8 dense taps (k=8 identity later)
    const float* fk = filt + k * NF * FPAD;

    // Filter operand pairs for this tap, cached in VGPRs. Same pairs serve
    // stage-1 B (row = s) and stage-2 A (row = c).
    v2f fslab[2][8];
#pragma unroll
    for (int rt = 0; rt < 2; ++rt)
#pragma unroll
      for (int q = 0; q < 8; ++q)
        fslab[rt][q] = *(const v2f*)&fk[(rt * 16 + m) * FPAD + 4 * q + 2 * hhi];

#pragma unroll
    for (int il = 0; il < IPW; ++il) {
      const int ii = w * IPW + il;
      const int tt = ii + k;

      // ---- stage-1 A operands: S_tt slabs (shared by both st-tiles) ----
      v2f aA[2][8];
#pragma unroll
      for (int Ct = 0; Ct < 2; ++Ct) {
        const float* ab = s_slab + ((2 * hhi) * NF + Ct * 16 + m) * SPAD + tt;
#pragma unroll
        for (int ks = 0; ks < 8; ++ks) {
          aA[Ct][ks].x = ab[(4 * ks) * NF * SPAD];
          aA[Ct][ks].y = ab[(4 * ks + 1) * NF * SPAD];
        }
      }

      // ---- stage 1: V = S_tt * F_k^T  -> transposed bounce [s][C] ----
#pragma unroll
      for (int Ct = 0; Ct < 2; ++Ct)
#pragma unroll
        for (int st = 0; st < 2; ++st) {
          v8f v = (v8f){0.f, 0.f, 0.f, 0.f, 0.f, 0.f, 0.f, 0.f};
#pragma unroll
          for (int ks = 0; ks < 8; ++ks)
            v = WMMA_F32X4(aA[Ct][ks], fslab[st][ks], v);
          float* vb = vstg + (st * 16 + m) * VPAD2 + Ct * 16 + hhi * 8;
#pragma unroll
          for (int r = 0; r < 8; ++r)
            vb[r] = v[r];
        }

      // ---- stage-2 B operands: contiguous V pairs (b64 loads) ----
      v2f vB[2][8];
#pragma unroll
      for (int st = 0; st < 2; ++st)
#pragma unroll
        for (int kc = 0; kc < 8; ++kc)
          vB[st][kc] = *(const v2f*)&vstg[(st * 16 + m) * VPAD2 + 4 * kc + 2 * hhi];

      // ---- stage 2: acc += F_k * V ----
#pragma unroll
      for (int ct = 0; ct < 2; ++ct)
#pragma unroll
        for (int st = 0; st < 2; ++st) {
          v8f o = acc[il][ct * 2 + st];
#pragma unroll
          for (int kc = 0; kc < 8; ++kc)
            o = WMMA_F32X4(fslab[ct][kc], vB[st][kc], o);
          acc[il][ct * 2 + st] = o;
        }
    }
  }

  // ---- identity tap: acc += S_{t+8} (base + r*SPAD immediate offsets) ----
#pragma unroll
  for (int il = 0; il < IPW; ++il) {
    const int ii = w * IPW + il;
#pragma unroll
    for (int ct = 0; ct < 2; ++ct)
#pragma unroll
      for (int st = 0; st < 2; ++st) {
        v8f o = acc[il][ct * 2 + st];
        const float* ib =
            s_slab + ((st * 16 + m) * NF + ct * 16 + hhi * 8) * SPAD + ii + ORDER;
#pragma unroll
        for (int r = 0; r < 8; ++r)
          o[r] += ib[r * SPAD];
        acc[il][ct * 2 + st] = o;
      }
  }

  __syncthreads();   // all waves done reading s_slab -> reuse as packed staging
  float* stage = s_slab;                           // [(c*32+s)*OPAD + ii], stride 24 (packed)
#pragma unroll
  for (int il = 0; il < IPW; ++il) {
    const int ii = w * IPW + il;
#pragma unroll
    for (int ct = 0; ct < 2; ++ct)
#pragma unroll
      for (int st = 0; st < 2; ++st) {
        v8f o = acc[il][ct * 2 + st];
        float* sb = stage + ((ct * 16 + hhi * 8) * NF + st * 16 + m) * OPAD + ii;
#pragma unroll
        for (int r = 0; r < 8; ++r)
          sb[r * NF * OPAD] = o[r];                // r*3072B -> DS immediate offsets
      }
  }
  __syncthreads();

  // ---- TDM store: 1024 x 24 packed tile -> x_Sigma (row stride 2040) ----
  if (w == 0) {
    float* gout = xSig + (size_t)b * NF * NF * TOUT + t0;
    tdm_issue<true>((uint64_t)(uintptr_t)gout, lds0 + SLAB_BYTE_OFF,
                    /*tdim0*/ TOUT, /*tdim1*/ NF * NF, /*str0*/ (uint64_t)TOUT,
                    /*tile0*/ TT, /*tile1*/ NF * NF,
                    /*pad*/ false, 0u, 0u, /*iterate*/ false, 0u, 0u, 0u);
    __builtin_amdgcn_s_wait_tensorcnt(0);          // LDS must stay allocated until done
  }
}

// ---------------------------------------------------------------------------
// x_mu kernel: tiny FLOP count; TDM-fed LDS tiles + plain VALU.
// ---------------------------------------------------------------------------
constexpr int TI = 120;              // 2040 = 17 * 120
constexpr int MU_ROW = TI + ORDER + 1;                   // 129 (TDM pad engine)
constexpr int MU_AMU_FLOATS = NF * MU_ROW;               // 4128
constexpr int MU_LDS_FLOATS = MU_AMU_FLOATS + NF * NF * ORDER;  // + 8192
constexpr size_t MU_LDS_BYTES = (size_t)MU_LDS_FLOATS * sizeof(float);
static_assert(TOUT % TI == 0, "mu tiling exact");

__global__ __launch_bounds__(256)
void mu_kernel(const float* __restrict__ a_mu, const float* __restrict__ ma,
               const float* __restrict__ bias, float* __restrict__ xMu) {
  extern __shared__ float mlds[];
  float* amu = mlds;                       // [C*129 + tt]
  float* mal = mlds + MU_AMU_FLOATS;       // filtr dense taps, flat [c][C][k]
  const int tid = threadIdx.x;
  const int t0  = blockIdx.x * TI;
  const int b   = blockIdx.y;
  const unsigned lds0 = __builtin_amdgcn_groupstaticsize();

  if ((tid >> 5) == 0) {
    // a_mu tile: x = 128 t, y = 32 C-rows (stride 2048); pad +1 DW / 128 DW -> 129 rows
    tdm_issue<false>((uint64_t)(uintptr_t)(a_mu + (size_t)b * NF * TIN + t0), lds0,
                     /*tdim0*/ TIN, /*tdim1*/ NF, /*str0*/ (uint64_t)TIN,
                     /*tile0*/ TI + ORDER, /*tile1*/ NF,
                     /*pad*/ true, /*ivl: 2^(6+1)=128 DW*/ 6u, /*amt: 1 DW*/ 0u,
                     /*iterate*/ false, 0u, 0u, 0u);
    // tap table: flat 8192-element 1D load
    tdm_issue<false>((uint64_t)(uintptr_t)ma, lds0 + MU_AMU_FLOATS * 4u,
                     /*tdim0*/ NF * NF * ORDER, /*tdim1*/ 1, /*str0*/ NF * NF * ORDER,
                     /*tile0*/ NF * NF * ORDER, /*tile1*/ 0,
                     /*pad*/ false, 0u, 0u, /*iterate*/ false, 0u, 0u, 0u);
    __builtin_amdgcn_s_wait_tensorcnt(0);
  }
  __syncthreads();

  for (int idx = tid; idx < NF * TI; idx += 256) {
    const int c = idx / TI;
    const int i = idx % TI;
    float sum = bias[c] + amu[c * MU_ROW + i + ORDER];   // identity tap
    for (int C = 0; C < NF; ++C) {
      const float* arow = &amu[C * MU_ROW + i];
      const float* f    = &mal[(c * NF + C) * ORDER];
#pragma unroll
      for (int k = 0; k < ORDER; ++k) sum += arow[k] * f[k];
    }
    xMu[((size_t)b * NF + c) * TOUT + t0 + i] = sum;
  }
}

// ---------------------------------------------------------------------------
extern "C" void kernel_launch(void* const* d_in, const int* in_sizes, int n_in,
                              void* d_out, int out_size, void* d_ws, size_t ws_size,
                              hipStream_t stream) {
  const float* a_mu    = (const float*)d_in[0];   // (16,32,2048)
  const float* a_Sigma = (const float*)d_in[1];   // (16,32,32,2048)
  const float* ma      = (const float*)d_in[2];   // (32,32,8)
  const float* bias    = (const float*)d_in[3];   // (32,)

  float* x_mu    = (float*)d_out;                          // (16,32,2040)
  float* x_Sigma = x_mu + (size_t)BATCH * NF * TOUT;       // (16,32,32,2040)

  (void)hipFuncSetAttribute(reinterpret_cast<const void*>(mu_kernel),
                            hipFuncAttributeMaxDynamicSharedMemorySize,
                            (int)MU_LDS_BYTES);
  mu_kernel<<<dim3(TOUT / TI, BATCH), 256, MU_LDS_BYTES, stream>>>(a_mu, ma, bias, x_mu);

  (void)hipFuncSetAttribute(reinterpret_cast<const void*>(sigma_wmma_kernel),
                            hipFuncAttributeMaxDynamicSharedMemorySize,
                            (int)LDS_BYTES);
  sigma_wmma_kernel<<<dim3(TOUT / TT, BATCH), NW * 32, LDS_BYTES, stream>>>(
      a_Sigma, ma, x_Sigma);
}